// JetTagger_72619307041455
// MI455X (gfx1250) — compile-verified
//
#include <hip/hip_runtime.h>
#include <math.h>
#include <stdint.h>

// ---------------------------------------------------------------------------
// GCN forward for MI455X (gfx1250, wave32).
// x[N,16] -> GCNConv(W1) -> relu -> GCNConv(W2) -> relu -> mean-pool[G,128]
// -> sigmoid(pooled@Wlab) and relu(pooled@Wd1)@Wd2 heads.
//
// GEMMs use V_WMMA_F32_16X16X4_F32 (exact fp32) with the A tile staged into
// LDS via GLOBAL_LOAD_ASYNC_TO_LDS_B128 (ASYNCcnt-tracked) when available.
// Edge aggregation is edge-parallel (1 wave / edge) with f32 global atomics;
// both N*128 fp32 feature buffers (25.6MB each) are L2-resident in 192MB L2.
// ---------------------------------------------------------------------------

typedef __attribute__((ext_vector_type(2))) float v2f;
typedef __attribute__((ext_vector_type(8))) float v8f;
typedef __attribute__((ext_vector_type(4))) int v4i;

#define HDIM 128
#define IN_DIM 16
#define G_GRAPHS 512

#ifndef __has_builtin
#define __has_builtin(x) 0
#endif

#if __has_builtin(__builtin_amdgcn_global_load_async_to_lds_b128)
#define HAVE_ASYNC_LDS 1
#else
#define HAVE_ASYNC_LDS 0
#endif

// addrspace-qualified int4 for the async-to-LDS builtin:
//   param0: addrspace(1) ("__device__") v4i*   (global source)
//   param1: addrspace(3) ("__shared__") v4i*   (LDS destination)
typedef __attribute__((address_space(1))) v4i as1_v4i;
typedef __attribute__((address_space(3))) v4i as3_v4i;

// ----------------------------- small utilities -----------------------------

__global__ void k_fill(float* __restrict__ p, float v, int n) {
  int i = blockIdx.x * blockDim.x + threadIdx.x;
  if (i < n) p[i] = v;
}

__global__ void k_deg_edges(float* __restrict__ deg, const int* __restrict__ dst,
                            int E) {
  int e = blockIdx.x * blockDim.x + threadIdx.x;
  if (e < E) atomicAdd(&deg[dst[e]], 1.0f);
}

__global__ void k_dinv(float* __restrict__ deg, int n) {
  int i = blockIdx.x * blockDim.x + threadIdx.x;
  if (i < n) {
    float d = deg[i];
    deg[i] = (d > 0.0f) ? rsqrtf(d) : 0.0f;
  }
}

// ------------------------------ WMMA GEMM ----------------------------------
// Out[N,128] = A[N,KDIM] @ W[KDIM,128], fp32 exact via v_wmma_f32_16x16x4_f32.
// Block = 256 threads = 8 waves. Block b owns rows [16b,16b+16); wave w owns
// cols [16w,16w+16). K stepped by 4 per WMMA. A tile staged in LDS (padded
// row stride to dodge bank conflicts on the stride-KDIM fragment reads).
//
// f32 16x16x4 fragment layouts (ISA 7.12.2):
//   A (16x4): lane l<16 -> row l, {K0,K1}; lane>=16 -> row l-16, {K2,K3}.
//   B (4x16): VGPR0 = {K0 row | K2 row}, VGPR1 = {K1 row | K3 row}, N=lane%16.
//   C/D (16x16): VGPR r -> row r (lanes 0-15) / row r+8 (lanes 16-31).

template <int KDIM>
__global__ __launch_bounds__(256) void k_gemm_wmma(
    const float* __restrict__ A, const float* __restrict__ W,
    float* __restrict__ Out, int Nrows) {
  constexpr int LSTR = KDIM + 4;           // padded LDS row stride (floats)
  __shared__ float As[16 * LSTR];

  const int tid = threadIdx.x;
  const int lane = tid & 31;
  const int wave = tid >> 5;
  const int m0 = blockIdx.x << 4;
  const int n0 = wave << 4;
  const int half = lane >> 4;  // 0 or 1
  const int l = lane & 15;

  // ---- stage A tile (16 x KDIM) into LDS, 16B chunks ----
  constexpr int CPR = KDIM / 4;            // float4 chunks per row
  constexpr int CHUNKS = 16 * CPR;         // 64 (K=16) or 512 (K=128)
  for (int c = tid; c < CHUNKS; c += 256) {
    const int r = c / CPR;
    const int cc = c % CPR;
    int row = m0 + r;
    if (row >= Nrows) row = Nrows - 1;     // clamp; OOB rows never stored
    const float* gp = A + (size_t)row * KDIM + cc * 4;
    float* lp = &As[r * LSTR + cc * 4];
#if HAVE_ASYNC_LDS
    __builtin_amdgcn_global_load_async_to_lds_b128(
        (as1_v4i*)(uintptr_t)gp, (as3_v4i*)lp, 0, 0);
#else
    *(float4*)lp = *(const float4*)gp;
#endif
  }
#if HAVE_ASYNC_LDS
#if __has_builtin(__builtin_amdgcn_s_wait_asynccnt)
  __builtin_amdgcn_s_wait_asynccnt(0);
#else
  asm volatile("s_wait_asynccnt 0x0" ::: "memory");
#endif
#endif
  __syncthreads();

  const float* arow = As + l * LSTR;       // fragment row for this lane
  v8f acc = {};
#pragma unroll
  for (int k0 = 0; k0 < KDIM; k0 += 4) {
    const int ka = k0 + half * 2;
    v2f a, b;
    a.x = arow[ka];
    a.y = arow[ka + 1];
    const int col = n0 + l;
    b.x = W[(size_t)(ka + 0) * HDIM + col];
    b.y = W[(size_t)(ka + 1) * HDIM + col];
    // 8 args: (neg_a, A, neg_b, B, c_mod, C, reuse_a, reuse_b)
    acc = __builtin_amdgcn_wmma_f32_16x16x4_f32(false, a, false, b, (short)0,
                                                acc, false, false);
  }

  const int rbase = m0 + half * 8;
  float* op = Out + (size_t)rbase * HDIM + n0 + l;
  if (m0 + 16 <= Nrows) {                  // uniform fast path: full tile
#pragma unroll
    for (int r = 0; r < 8; ++r) op[(size_t)r * HDIM] = acc[r];
  } else {                                 // tail tile (unused for N=50000)
#pragma unroll
    for (int r = 0; r < 8; ++r)
      if (rbase + r < Nrows) op[(size_t)r * HDIM] = acc[r];
  }
}

// --------------------------- GCN aggregation -------------------------------
// out[i,:] = dinv[i]^2 * h[i,:]         (self-loop term, also initializes out)
__global__ void k_self_init(const float* __restrict__ h, float* __restrict__ out,
                            const float* __restrict__ dinv, int N) {
  int gid = blockIdx.x * blockDim.x + threadIdx.x;
  int i = gid >> 5;
  if (i >= N) return;
  int lane = gid & 31;
  float di = dinv[i];
  float nrm = di * di;
  float4 v = ((const float4*)(h + (size_t)i * HDIM))[lane];
  float4 o;
  o.x = v.x * nrm; o.y = v.y * nrm; o.z = v.z * nrm; o.w = v.w * nrm;
  ((float4*)(out + (size_t)i * HDIM))[lane] = o;
}

// out[dst,:] += dinv[src]*dinv[dst] * h[src,:]   (one wave32 per edge)
__global__ void k_agg_edges(const float* __restrict__ h, float* __restrict__ out,
                            const float* __restrict__ dinv,
                            const int* __restrict__ src,
                            const int* __restrict__ dst, int E) {
  int gid = blockIdx.x * blockDim.x + threadIdx.x;
  int e = gid >> 5;
  if (e >= E) return;
  int lane = gid & 31;
  int s = src[e];
  int d = dst[e];
  float nrm = dinv[s] * dinv[d];
  float4 v = ((const float4*)(h + (size_t)s * HDIM))[lane];
  float* op = out + (size_t)d * HDIM + lane * 4;
  atomicAdd(op + 0, v.x * nrm);
  atomicAdd(op + 1, v.y * nrm);
  atomicAdd(op + 2, v.z * nrm);
  atomicAdd(op + 3, v.w * nrm);
}

// out = relu(in + b[f])
__global__ void k_relu_bias(const float* __restrict__ in,
                            const float* __restrict__ b,
                            float* __restrict__ out, long long total) {
  long long idx = (long long)blockIdx.x * blockDim.x + threadIdx.x;
  if (idx >= total) return;
  int f = (int)(idx & (HDIM - 1));
  out[idx] = fmaxf(in[idx] + b[f], 0.0f);
}

// ------------------------------- pooling -----------------------------------
__global__ void k_pool(const float* __restrict__ h, const int* __restrict__ batch,
                       float* __restrict__ sums, float* __restrict__ cnts,
                       int N) {
  int gid = blockIdx.x * blockDim.x + threadIdx.x;
  int i = gid >> 5;
  if (i >= N) return;
  int lane = gid & 31;
  int g = batch[i];
  float4 v = ((const float4*)(h + (size_t)i * HDIM))[lane];
  float* sp = sums + (size_t)g * HDIM + lane * 4;
  atomicAdd(sp + 0, v.x);
  atomicAdd(sp + 1, v.y);
  atomicAdd(sp + 2, v.z);
  atomicAdd(sp + 3, v.w);
  if (lane == 0) atomicAdd(&cnts[g], 1.0f);
}

// ------------------------------- heads -------------------------------------
// One block of 128 threads per graph.
__global__ void k_heads(const float* __restrict__ sums,
                        const float* __restrict__ cnts,
                        const float* __restrict__ Wlab,
                        const float* __restrict__ blab,
                        const float* __restrict__ Wd1,
                        const float* __restrict__ bd1,
                        const float* __restrict__ Wd2,
                        const float* __restrict__ bd2,
                        float* __restrict__ out) {
  __shared__ float pooled[HDIM];
  __shared__ float hidden[64];
  const int g = blockIdx.x;
  const int t = threadIdx.x;

  float c = fmaxf(cnts[g], 1.0f);
  pooled[t] = sums[(size_t)g * HDIM + t] / c;
  __syncthreads();

  if (t < 64) {
    float acc = bd1[t];
#pragma unroll 4
    for (int k = 0; k < HDIM; ++k) acc += pooled[k] * Wd1[k * 64 + t];
    hidden[t] = fmaxf(acc, 0.0f);
  }
  __syncthreads();

  if (t == 0) {
    float acc = blab[0];
#pragma unroll 4
    for (int k = 0; k < HDIM; ++k) acc += pooled[k] * Wlab[k];
    out[g] = 1.0f / (1.0f + __expf(-acc));  // label_out[g,0]
  }
  if (t < 2) {
    float acc = bd2[t];
#pragma unroll 4
    for (int k = 0; k < 64; ++k) acc += hidden[k] * Wd2[k * 2 + t];
    out[G_GRAPHS + g * 2 + t] = acc;        // dom_out[g,t]
  }
}

// ------------------------------- launcher ----------------------------------

extern "C" void kernel_launch(void* const* d_in, const int* in_sizes, int n_in,
                              void* d_out, int out_size, void* d_ws,
                              size_t ws_size, hipStream_t stream) {
  const float* x     = (const float*)d_in[0];
  const int*   ei    = (const int*)d_in[1];
  const int*   batch = (const int*)d_in[2];
  const float* W1    = (const float*)d_in[3];
  const float* b1    = (const float*)d_in[4];
  const float* W2    = (const float*)d_in[5];
  const float* b2    = (const float*)d_in[6];
  const float* Wlab  = (const float*)d_in[7];
  const float* blab  = (const float*)d_in[8];
  const float* Wd1   = (const float*)d_in[9];
  const float* bd1   = (const float*)d_in[10];
  const float* Wd2   = (const float*)d_in[11];
  const float* bd2   = (const float*)d_in[12];

  const int N = in_sizes[0] / IN_DIM;   // 50000
  const int E = in_sizes[1] / 2;        // 1600000
  const int* src = ei;
  const int* dst = ei + E;
  float* out = (float*)d_out;

  // Workspace carve-up (all fp32). N*4 = 200000 bytes keeps 16B alignment.
  float* deg  = (float*)d_ws;                    // [N]  degree -> dinv (inplace)
  float* bufA = deg + N;                          // [N*128]
  float* bufB = bufA + (size_t)N * HDIM;          // [N*128]
  float* sums = bufB + (size_t)N * HDIM;          // [G*128]
  float* cnts = sums + (size_t)G_GRAPHS * HDIM;   // [G] (contiguous with sums)

  const int T = 256;
  auto cdiv = [](long long a, long long b) { return (int)((a + b - 1) / b); };

  // --- degree (self-loop counts as 1) -> dinv ---
  k_fill<<<cdiv(N, T), T, 0, stream>>>(deg, 1.0f, N);
  k_deg_edges<<<cdiv(E, T), T, 0, stream>>>(deg, dst, E);
  k_dinv<<<cdiv(N, T), T, 0, stream>>>(deg, N);

  // --- layer 1: h1 = relu(Agg(x @ W1) + b1) ---
  k_gemm_wmma<IN_DIM><<<cdiv(N, 16), T, 0, stream>>>(x, W1, bufA, N);
  k_self_init<<<cdiv((long long)N * 32, T), T, 0, stream>>>(bufA, bufB, deg, N);
  k_agg_edges<<<cdiv((long long)E * 32, T), T, 0, stream>>>(bufA, bufB, deg,
                                                            src, dst, E);
  k_relu_bias<<<cdiv((long long)N * HDIM, T), T, 0, stream>>>(
      bufB, b1, bufA, (long long)N * HDIM);

  // --- layer 2: h2 = relu(Agg(h1 @ W2) + b2) ---
  k_gemm_wmma<HDIM><<<cdiv(N, 16), T, 0, stream>>>(bufA, W2, bufB, N);
  k_self_init<<<cdiv((long long)N * 32, T), T, 0, stream>>>(bufB, bufA, deg, N);
  k_agg_edges<<<cdiv((long long)E * 32, T), T, 0, stream>>>(bufB, bufA, deg,
                                                            src, dst, E);
  k_relu_bias<<<cdiv((long long)N * HDIM, T), T, 0, stream>>>(
      bufA, b2, bufA, (long long)N * HDIM);

  // --- global mean pool + heads ---
  k_fill<<<cdiv(G_GRAPHS * HDIM + G_GRAPHS, T), T, 0, stream>>>(
      sums, 0.0f, G_GRAPHS * HDIM + G_GRAPHS);
  k_pool<<<cdiv((long long)N * 32, T), T, 0, stream>>>(bufA, batch, sums, cnts, N);
  k_heads<<<G_GRAPHS, HDIM, 0, stream>>>(sums, cnts, Wlab, blab, Wd1, bd1,
                                         Wd2, bd2, out);
}